// ActionValues_65403761984233
// MI455X (gfx1250) — compile-verified
//
#include <hip/hip_runtime.h>
#include <math.h>

// ---- model constants ----
#define KA   12          // agents
#define MOA  12
#define HW   576         // 24*24
#define C0   32
#define D0   8
#define C1   16
#define C2   16
#define C3   8
#define HSZ  4608        // C3*576
#define INP  5024        // HSZ + 16 + 16 + 12*16 + 12*16
#define G4   18432       // 4*HSZ
#define EPSV 1e-5f

typedef __attribute__((ext_vector_type(2))) float v2f;
typedef __attribute__((ext_vector_type(8))) float v8f;

// ---------------- conv1: einsum bcdhw,ocd->bohw + bias ----------------
__global__ void conv1_kernel(const float* __restrict__ x, const float* __restrict__ w,
                             const float* __restrict__ b, float* __restrict__ out) {
  const int k = blockIdx.x, o = blockIdx.y;
  for (int pix = threadIdx.x; pix < HW; pix += blockDim.x) {
    float acc = b[o];
    const float* xp = x + (size_t)k * C0 * D0 * HW + pix;
    const float* wp = w + (size_t)o * C0 * D0;
#pragma unroll 8
    for (int cd = 0; cd < C0 * D0; ++cd)
      acc = fmaf(xp[(size_t)cd * HW], wp[cd], acc);
    out[((size_t)k * C1 + o) * HW + pix] = acc;
  }
}

// ------------- batchnorm (training stats over N,H,W) + relu -------------
// in: [12, C, 576]; out row stride per agent = out_stride (elements)
__global__ void bn_relu_kernel(const float* __restrict__ in, const float* __restrict__ g,
                               const float* __restrict__ bb, float* __restrict__ out,
                               int C, int out_stride) {
  const int c = blockIdx.x;
  __shared__ float s1[256], s2[256];
  float sum = 0.f, sq = 0.f;
  for (int i = threadIdx.x; i < KA * HW; i += blockDim.x) {
    int k = i / HW, pix = i % HW;
    float v = in[((size_t)k * C + c) * HW + pix];
    sum += v; sq += v * v;
  }
  s1[threadIdx.x] = sum; s2[threadIdx.x] = sq;
  __syncthreads();
  for (int off = 128; off; off >>= 1) {
    if (threadIdx.x < off) { s1[threadIdx.x] += s1[threadIdx.x + off]; s2[threadIdx.x] += s2[threadIdx.x + off]; }
    __syncthreads();
  }
  const float inv_n = 1.f / (float)(KA * HW);
  float mu  = s1[0] * inv_n;
  float var = s2[0] * inv_n - mu * mu;
  float scale = g[c] * rsqrtf(var + EPSV);
  float shift = bb[c] - mu * scale;
  for (int i = threadIdx.x; i < KA * HW; i += blockDim.x) {
    int k = i / HW, pix = i % HW;
    float v = in[((size_t)k * C + c) * HW + pix] * scale + shift;
    out[(size_t)k * out_stride + (size_t)c * HW + pix] = fmaxf(v, 0.f);
  }
}

// ---------------- conv2: 3x3 SAME, 16->16 ----------------
__global__ void conv2_kernel(const float* __restrict__ h1, const float* __restrict__ w,
                             const float* __restrict__ b, float* __restrict__ out) {
  const int k = blockIdx.x, o = blockIdx.y;
  for (int pix = threadIdx.x; pix < HW; pix += blockDim.x) {
    int y = pix / 24, xx = pix % 24;
    float acc = b[o];
    for (int ci = 0; ci < C1; ++ci) {
      const float* hp = h1 + ((size_t)k * C1 + ci) * HW;
      const float* wp = w + ((size_t)o * C1 + ci) * 9;
#pragma unroll
      for (int dy = 0; dy < 3; ++dy) {
        int yy = y + dy - 1; if ((unsigned)yy >= 24u) continue;
#pragma unroll
        for (int dx = 0; dx < 3; ++dx) {
          int xc = xx + dx - 1; if ((unsigned)xc >= 24u) continue;
          acc = fmaf(hp[yy * 24 + xc], wp[dy * 3 + dx], acc);
        }
      }
    }
    out[((size_t)k * C2 + o) * HW + pix] = acc;
  }
}

// ---------------- conv3: 5x5 SAME, 16->8 ----------------
__global__ void conv3_kernel(const float* __restrict__ h2, const float* __restrict__ w,
                             const float* __restrict__ b, float* __restrict__ out) {
  const int k = blockIdx.x, o = blockIdx.y;
  for (int pix = threadIdx.x; pix < HW; pix += blockDim.x) {
    int y = pix / 24, xx = pix % 24;
    float acc = b[o];
    for (int ci = 0; ci < C2; ++ci) {
      const float* hp = h2 + ((size_t)k * C2 + ci) * HW;
      const float* wp = w + ((size_t)o * C2 + ci) * 25;
#pragma unroll
      for (int dy = 0; dy < 5; ++dy) {
        int yy = y + dy - 2; if ((unsigned)yy >= 24u) continue;
#pragma unroll
        for (int dx = 0; dx < 5; ++dx) {
          int xc = xx + dx - 2; if ((unsigned)xc >= 24u) continue;
          acc = fmaf(hp[yy * 24 + xc], wp[dy * 5 + dx], acc);
        }
      }
    }
    out[((size_t)k * C3 + o) * HW + pix] = acc;
  }
}

// ------- phys/ment MLPs + assemble non-conv part of lstm_in, zero-pad rows 12..15 -------
__global__ void assemble_kernel(const float* __restrict__ p, const float* __restrict__ m,
                                const float* __restrict__ vis,
                                const float* __restrict__ phys_w, const float* __restrict__ phys_b,
                                const float* __restrict__ ment_w, const float* __restrict__ ment_b,
                                float* __restrict__ lstm_in /* [16][INP] */) {
  __shared__ float op[12][16];  // relu(p @ phys_w.T + b)
  __shared__ float mm[12][16];  // relu(0.9*m @ ment_w.T + b)   (mixed mental)
  __shared__ float am[12][16];  // relu(m @ ment_w.T + b)       (agent mental)
  const int t = threadIdx.x;
  if (t < 192) {
    int j = t >> 4, i = t & 15;
    float a = phys_b[i], bm = ment_b[i], cm = ment_b[i];
    for (int q = 0; q < 16; ++q) {
      a  = fmaf(p[j * 16 + q], phys_w[i * 16 + q], a);
      bm = fmaf(0.9f * m[j * 16 + q], ment_w[i * 16 + q], bm);
      cm = fmaf(m[j * 16 + q], ment_w[i * 16 + q], cm);
    }
    op[j][i] = fmaxf(a, 0.f); mm[j][i] = fmaxf(bm, 0.f); am[j][i] = fmaxf(cm, 0.f);
  }
  __syncthreads();
  // zero pad rows 12..15 (WMMA M padding)
  for (int idx = t; idx < 4 * INP; idx += blockDim.x) lstm_in[(size_t)12 * INP + idx] = 0.f;
  // tail of each real row: [ag(32) | phys_inp(192) | ment_inp(192)]
  for (int idx = t; idx < 12 * 416; idx += blockDim.x) {
    int k = idx / 416, r = idx % 416;
    float v;
    if (r < 16)        v = op[k][r];
    else if (r < 32)   v = am[k][r - 16];
    else if (r < 224) { int j = (r - 32) >> 4, i = (r - 32) & 15; v = vis[k * MOA + j] * op[j][i]; }
    else              { int j = (r - 224) >> 4, i = (r - 224) & 15; v = mm[j][i]; }
    lstm_in[(size_t)k * INP + HSZ + r] = v;
  }
}

// ------------- gates[12,18432] = lstm_in @ w_ih.T + b_ih + b_hh via FP32 WMMA -------------
// One wave per 16-row tile of w_ih; D = A(16xK) x B(Kx16), K stepped by 4 per wmma.
__global__ void __launch_bounds__(128) gates_wmma_kernel(
    const float* __restrict__ A,   // [16][INP] zero-padded lstm_in
    const float* __restrict__ W,   // [G4][INP] w_ih
    const float* __restrict__ b_ih, const float* __restrict__ b_hh,
    float* __restrict__ gates)     // [12][G4]
{
  const int lane  = threadIdx.x & 31;
  const int wave  = threadIdx.x >> 5;
  const int ntile = blockIdx.x * 4 + wave;         // 0..1151
  const int n     = ntile * 16 + (lane & 15);
  const int koff  = (lane >> 4) << 1;              // 0 (lanes 0-15) or 2 (lanes 16-31)
  const float* ap = A + (size_t)(lane & 15) * INP + koff;
  const float* wp = W + (size_t)n * INP + koff;
  v8f acc = {};
  for (int kk = 0; kk < INP; kk += 16) {           // 314 iterations, 4 wmma each
    __builtin_prefetch(wp + kk + 256, 0, 3);       // -> global_prefetch_b8, ~1KB ahead
    v2f a0 = *(const v2f*)(ap + kk);
    v2f b0 = *(const v2f*)(wp + kk);
    v2f a1 = *(const v2f*)(ap + kk + 4);
    v2f b1 = *(const v2f*)(wp + kk + 4);
    v2f a2 = *(const v2f*)(ap + kk + 8);
    v2f b2 = *(const v2f*)(wp + kk + 8);
    v2f a3 = *(const v2f*)(ap + kk + 12);
    v2f b3 = *(const v2f*)(wp + kk + 12);
    acc = __builtin_amdgcn_wmma_f32_16x16x4_f32(false, a0, false, b0, (short)0, acc, false, false);
    acc = __builtin_amdgcn_wmma_f32_16x16x4_f32(false, a1, false, b1, (short)0, acc, false, false);
    acc = __builtin_amdgcn_wmma_f32_16x16x4_f32(false, a2, false, b2, (short)0, acc, false, false);
    acc = __builtin_amdgcn_wmma_f32_16x16x4_f32(false, a3, false, b3, (short)0, acc, false, false);
  }
  const float bias = b_ih[n] + b_hh[n];
#pragma unroll
  for (int i = 0; i < 8; ++i) {
    int mrow = i + ((lane >> 4) << 3);             // C/D layout: vgpr i -> M=i (lanes<16), M=i+8 (lanes>=16)
    if (mrow < 12) gates[(size_t)mrow * G4 + n] = acc[i] + bias;
  }
}

// ---------------- LSTM elementwise (c0 = h0 = 0) + relu ----------------
__global__ void lstm_kernel(const float* __restrict__ gates, float* __restrict__ hr) {
  int idx = blockIdx.x * blockDim.x + threadIdx.x;
  if (idx >= KA * HSZ) return;
  int k = idx / HSZ, j = idx % HSZ;
  const float* gr = gates + (size_t)k * G4;
  float ig = gr[j], gg = gr[2 * HSZ + j], og = gr[3 * HSZ + j];  // f-gate * c0 == 0
  float c = (1.f / (1.f + expf(-ig))) * tanhf(gg);
  float h = (1.f / (1.f + expf(-og))) * tanhf(c);
  hr[idx] = fmaxf(h, 0.f);
}

// ---------------- out[k][n] = (relu?) dot(hr[k], W[n]) + b[n] ----------------
__global__ void matvec_kernel(const float* __restrict__ hr, const float* __restrict__ W,
                              const float* __restrict__ b, float* __restrict__ out,
                              int N, int relu) {
  const int n = blockIdx.x, k = blockIdx.y;
  __shared__ float s[128];
  const float* h = hr + (size_t)k * HSZ;
  const float* w = W + (size_t)n * HSZ;
  float acc = 0.f;
  for (int i = threadIdx.x; i < HSZ; i += 128) acc = fmaf(h[i], w[i], acc);
  s[threadIdx.x] = acc;
  __syncthreads();
  for (int off = 64; off; off >>= 1) {
    if (threadIdx.x < off) s[threadIdx.x] += s[threadIdx.x + off];
    __syncthreads();
  }
  if (threadIdx.x == 0) {
    float v = s[0] + b[n];
    out[(size_t)k * N + n] = relu ? fmaxf(v, 0.f) : v;
  }
}

// ---------------- act[12,16] = ah_out[12,128] @ act_w.T + act_b ----------------
__global__ void act_kernel(const float* __restrict__ ah, const float* __restrict__ act_w,
                           const float* __restrict__ act_b, float* __restrict__ out) {
  int t = threadIdx.x;
  if (t >= 192) return;
  int k = t >> 4, a = t & 15;
  float acc = act_b[a];
#pragma unroll 4
  for (int q = 0; q < 128; ++q) acc = fmaf(ah[k * 128 + q], act_w[a * 128 + q], acc);
  out[k * 16 + a] = acc;
}

extern "C" void kernel_launch(void* const* d_in, const int* in_sizes, int n_in,
                              void* d_out, int out_size, void* d_ws, size_t ws_size,
                              hipStream_t stream) {
  (void)in_sizes; (void)n_in; (void)out_size; (void)ws_size;
  const float* x       = (const float*)d_in[0];
  const float* p       = (const float*)d_in[1];
  const float* m       = (const float*)d_in[2];
  const float* vis     = (const float*)d_in[3];
  const float* conv1_w = (const float*)d_in[4];
  const float* conv1_b = (const float*)d_in[5];
  const float* bn1_g   = (const float*)d_in[6];
  const float* bn1_b   = (const float*)d_in[7];
  const float* conv2_w = (const float*)d_in[8];
  const float* conv2_b = (const float*)d_in[9];
  const float* bn2_g   = (const float*)d_in[10];
  const float* bn2_b   = (const float*)d_in[11];
  const float* conv3_w = (const float*)d_in[12];
  const float* conv3_b = (const float*)d_in[13];
  const float* bn3_g   = (const float*)d_in[14];
  const float* bn3_b   = (const float*)d_in[15];
  const float* phys_w  = (const float*)d_in[16];
  const float* phys_b  = (const float*)d_in[17];
  const float* ment_w  = (const float*)d_in[18];
  const float* ment_b  = (const float*)d_in[19];
  const float* w_ih    = (const float*)d_in[20];
  // d_in[21] = w_hh: intentionally unused (h0 == 0) -> saves 340 MB of HBM traffic
  const float* b_ih    = (const float*)d_in[22];
  const float* b_hh    = (const float*)d_in[23];
  const float* ah_w    = (const float*)d_in[24];
  const float* ah_b    = (const float*)d_in[25];
  const float* act_w   = (const float*)d_in[26];
  const float* act_b   = (const float*)d_in[27];
  const float* inf_w   = (const float*)d_in[28];
  const float* inf_b   = (const float*)d_in[29];
  float* out = (float*)d_out;

  // workspace layout (floats)
  float* ws      = (float*)d_ws;
  float* h1pre   = ws;                      // 12*16*576 = 110592
  float* h1      = h1pre + 110592;          // 110592
  float* h2pre   = h1    + 110592;          // 110592
  float* h2      = h2pre + 110592;          // 110592
  float* h3pre   = h2    + 110592;          // 55296
  float* lstm_in = h3pre + 55296;           // 16*5024 = 80384 (rows 12..15 zero)
  float* gates   = lstm_in + 80384;         // 12*18432 = 221184
  float* hr      = gates + 221184;          // 12*4608 = 55296
  float* ahout   = hr    + 55296;           // 12*128

  // conv tower + batchnorm(+relu)
  conv1_kernel<<<dim3(KA, C1), 192, 0, stream>>>(x, conv1_w, conv1_b, h1pre);
  bn_relu_kernel<<<C1, 256, 0, stream>>>(h1pre, bn1_g, bn1_b, h1, C1, C1 * HW);
  conv2_kernel<<<dim3(KA, C2), 192, 0, stream>>>(h1, conv2_w, conv2_b, h2pre);
  bn_relu_kernel<<<C2, 256, 0, stream>>>(h2pre, bn2_g, bn2_b, h2, C2, C2 * HW);
  conv3_kernel<<<dim3(KA, C3), 192, 0, stream>>>(h2, conv3_w, conv3_b, h3pre);
  // write h3 directly into lstm_in[:, 0:4608]
  bn_relu_kernel<<<C3, 256, 0, stream>>>(h3pre, bn3_g, bn3_b, lstm_in, C3, INP);
  // MLPs + tail of lstm_in + zero M-padding rows
  assemble_kernel<<<1, 256, 0, stream>>>(p, m, vis, phys_w, phys_b, ment_w, ment_b, lstm_in);

  // big memory-bound GEMM on the WMMA pipe: 1152 N-tiles, 4 waves/block
  gates_wmma_kernel<<<288, 128, 0, stream>>>(lstm_in, w_ih, b_ih, b_hh, gates);

  // LSTM nonlinearity -> hr
  lstm_kernel<<<(KA * HSZ + 255) / 256, 256, 0, stream>>>(gates, hr);

  // heads
  matvec_kernel<<<dim3(128, KA), 128, 0, stream>>>(hr, ah_w, ah_b, ahout, 128, 1);
  act_kernel<<<1, 256, 0, stream>>>(ahout, act_w, act_b, out);                 // act -> out[0:192]
  matvec_kernel<<<dim3(192, KA), 128, 0, stream>>>(hr, inf_w, inf_b, out + 192, 192, 0); // old_ment
}